// Repro_21414706938483
// MI455X (gfx1250) — compile-verified
//
#include <hip/hip_runtime.h>

typedef __attribute__((ext_vector_type(2))) float v2f;
typedef __attribute__((ext_vector_type(8))) float v8f;

#define B_DIM 16
#define J_DIM 4
#define N_DIM 131072
#define BLOCKS_PER_BATCH 64
#define THREADS 256
#define PTS_PER_BLOCK (N_DIM / BLOCKS_PER_BATCH) // 2048
#define NACC 10

// One fused pass: skin + rigid transform + raw-moment accumulation.
// 4 points per thread -> all global traffic is 128-bit and 16B-aligned.
// moments_partial layout: [(b * BLOCKS_PER_BATCH + chunk) * NACC + q]
// q: 0..2 = sum m*w_xyz, 3..5 = sum m*w^2 diag, 6..8 = m*xy, m*xz, m*yz, 9 = sum m
__global__ __launch_bounds__(THREADS) void skin_moments_kernel(
    const float* __restrict__ angles,   // (B,J)
    const float* __restrict__ points,   // (B,N,3)
    const float* __restrict__ origins,  // (J,3)
    const float* __restrict__ joint_w,  // (J,B,N)
    const float* __restrict__ base_w,   // (B,N)
    const float* __restrict__ trans,    // (B,3)
    const float* __restrict__ rot,      // (B,3,3)
    const float* __restrict__ mass,     // (N)
    float* __restrict__ world,          // (B,N,3)
    float* __restrict__ moments_partial)
{
    const int b     = blockIdx.y;
    const int chunk = blockIdx.x;
    const int tid   = threadIdx.x;

    // Uniform per-batch constants -> scalar loads
    float cj[J_DIM], sj[J_DIM], ox[J_DIM], oy[J_DIM], oz[J_DIM];
#pragma unroll
    for (int j = 0; j < J_DIM; ++j) {
        float a = angles[b * J_DIM + j];
        cj[j] = __builtin_cosf(a);
        sj[j] = __builtin_sinf(a);
        ox[j] = origins[j * 3 + 0];
        oy[j] = origins[j * 3 + 1];
        oz[j] = origins[j * 3 + 2];
    }
    const float r00 = rot[b * 9 + 0], r01 = rot[b * 9 + 1], r02 = rot[b * 9 + 2];
    const float r10 = rot[b * 9 + 3], r11 = rot[b * 9 + 4], r12 = rot[b * 9 + 5];
    const float r20 = rot[b * 9 + 6], r21 = rot[b * 9 + 7], r22 = rot[b * 9 + 8];
    const float tx = trans[b * 3 + 0], ty = trans[b * 3 + 1], tz = trans[b * 3 + 2];

    const size_t bN   = (size_t)b * N_DIM;
    const size_t JW_S = (size_t)B_DIM * N_DIM; // stride between joint slices
    const int base    = chunk * PTS_PER_BLOCK;

    float acc[NACC];
#pragma unroll
    for (int q = 0; q < NACC; ++q) acc[q] = 0.f;

    for (int i0 = tid * 4; i0 < PTS_PER_BLOCK; i0 += THREADS * 4) {
        const int n0 = base + i0;           // multiple of 4
        const size_t idx0 = bN + n0;

        // ---- 128-bit loads, all 16B-aligned ----
        const float4* pp4 = (const float4*)(points + idx0 * 3);
        __builtin_prefetch(pp4 + (size_t)THREADS * 3, 0, 0);      // global_prefetch_b8
        __builtin_prefetch(joint_w + idx0 + THREADS * 4, 0, 0);
        const float4 f0 = pp4[0], f1 = pp4[1], f2 = pp4[2];

        const float4 bw4 = *(const float4*)(base_w + idx0);
        const float4 m4  = *(const float4*)(mass + n0);
        float4 w4[J_DIM];
#pragma unroll
        for (int j = 0; j < J_DIM; ++j)
            w4[j] = *(const float4*)(joint_w + (size_t)j * JW_S + idx0);

        // unpack AoS xyz for 4 points
        const float px[4] = { f0.x, f0.w, f1.z, f2.y };
        const float py[4] = { f0.y, f1.x, f1.w, f2.z };
        const float pz[4] = { f0.z, f1.y, f2.x, f2.w };
        const float bw[4] = { bw4.x, bw4.y, bw4.z, bw4.w };
        const float mm[4] = { m4.x,  m4.y,  m4.z,  m4.w  };

        float wxo[4], wyo[4], wzo[4];
#pragma unroll
        for (int k = 0; k < 4; ++k) {
            float bx = bw[k] * px[k], by = bw[k] * py[k], bz = bw[k] * pz[k];
#pragma unroll
            for (int j = 0; j < J_DIM; ++j) {
                const float w  = (j == 0) ? w4[0].x : (j == 1) ? w4[1].x
                                : (j == 2) ? w4[2].x : w4[3].x; // placeholder, fixed below
                (void)w;
            }
            // joint blend (explicit per-joint to index float4 lanes cleanly)
            const float wj[4] = { ((const float*)&w4[0])[k], ((const float*)&w4[1])[k],
                                  ((const float*)&w4[2])[k], ((const float*)&w4[3])[k] };
#pragma unroll
            for (int j = 0; j < J_DIM; ++j) {
                const float rx = px[k] - ox[j];
                const float ry = py[k] - oy[j];
                const float rz = pz[k] - oz[j];
                // moved_e = sum_d rel_d * R[d][e];  R = [[c,0,s],[0,1,0],[-s,0,c]]
                const float mx = rx * cj[j] - rz * sj[j] + ox[j];
                const float my = ry + oy[j];
                const float mz = rx * sj[j] + rz * cj[j] + oz[j];
                bx = fmaf(wj[j], mx, bx);
                by = fmaf(wj[j], my, by);
                bz = fmaf(wj[j], mz, bz);
            }
            // world_e = sum_d blended_d * rot[b][e][d] + trans[b][e]
            const float wx = bx * r00 + by * r01 + bz * r02 + tx;
            const float wy = bx * r10 + by * r11 + bz * r12 + ty;
            const float wz = bx * r20 + by * r21 + bz * r22 + tz;
            wxo[k] = wx; wyo[k] = wy; wzo[k] = wz;

            const float m = mm[k];
            acc[0] = fmaf(m, wx, acc[0]);
            acc[1] = fmaf(m, wy, acc[1]);
            acc[2] = fmaf(m, wz, acc[2]);
            acc[3] = fmaf(m * wx, wx, acc[3]);
            acc[4] = fmaf(m * wy, wy, acc[4]);
            acc[5] = fmaf(m * wz, wz, acc[5]);
            acc[6] = fmaf(m * wx, wy, acc[6]);
            acc[7] = fmaf(m * wx, wz, acc[7]);
            acc[8] = fmaf(m * wy, wz, acc[8]);
            acc[9] += m;
        }

        // ---- 128-bit stores ----
        float4* wp4 = (float4*)(world + idx0 * 3);
        wp4[0] = make_float4(wxo[0], wyo[0], wzo[0], wxo[1]);
        wp4[1] = make_float4(wyo[1], wzo[1], wxo[2], wyo[2]);
        wp4[2] = make_float4(wzo[2], wxo[3], wyo[3], wzo[3]);
    }

    // wave32 intra-wave reduction
#pragma unroll
    for (int q = 0; q < NACC; ++q) {
        float v = acc[q];
#pragma unroll
        for (int off = 16; off > 0; off >>= 1) v += __shfl_down(v, off, 32);
        acc[q] = v;
    }

    // Stage per-wave totals into LDS as a 16x8 tile (rows=accumulator, cols=wave).
    __shared__ float sums[16][8];
    if (tid < 128) ((float*)sums)[tid] = 0.f; // zero incl. pad rows 10..15
    __syncthreads();
    const int wave = tid >> 5;
    const int lane = tid & 31;
    if (lane == 0) {
#pragma unroll
        for (int q = 0; q < NACC; ++q) sums[q][wave] = acc[q];
    }
    __syncthreads();

    // Wave 0 (EXEC all ones): reduce the 8 wave totals per accumulator with
    // V_WMMA_F32_16X16X4_F32: C = ones(16x4) x B(4x16). Two WMMAs cover waves
    // 0..3 and 4..7; C row M=0 (lanes 0..15, vgpr 0) holds the column sums.
    if (tid < 32) {
        const int ncol = lane & 15;
        const int k0   = (lane < 16) ? 0 : 2; // B layout: v0 = rows {0,2}, v1 = rows {1,3}
        v2f a;  a.x = 1.f; a.y = 1.f;        // A[m][k] = 1 for all m,k
        v2f b0; b0.x = sums[ncol][k0];     b0.y = sums[ncol][k0 + 1];
        v2f b1; b1.x = sums[ncol][k0 + 4]; b1.y = sums[ncol][k0 + 5];
        v8f cacc = {};
        cacc = __builtin_amdgcn_wmma_f32_16x16x4_f32(false, a, false, b0,
                                                     (short)0, cacc, false, false);
        cacc = __builtin_amdgcn_wmma_f32_16x16x4_f32(false, a, false, b1,
                                                     (short)0, cacc, false, false);
        if (lane < NACC)
            moments_partial[((size_t)b * BLOCKS_PER_BATCH + chunk) * NACC + lane] = cacc[0];
    }
}

__global__ void finalize_kernel(const float* __restrict__ moments_partial,
                                float* __restrict__ out_com,
                                float* __restrict__ out_inertia)
{
    const int b = threadIdx.x;
    if (b >= B_DIM) return;

    float Mo[NACC];
#pragma unroll
    for (int q = 0; q < NACC; ++q) Mo[q] = 0.f;
    for (int cidx = 0; cidx < BLOCKS_PER_BATCH; ++cidx) {
        const float* p = moments_partial + ((size_t)b * BLOCKS_PER_BATCH + cidx) * NACC;
#pragma unroll
        for (int q = 0; q < NACC; ++q) Mo[q] += p[q];
    }

    const float M  = Mo[9];
    const float cx = Mo[0] / M, cy = Mo[1] / M, cz = Mo[2] / M;
    out_com[b * 3 + 0] = cx;
    out_com[b * 3 + 1] = cy;
    out_com[b * 3 + 2] = cz;

    // central second moments: C_ab = S2_ab - M * c_a * c_b
    const float Cxx = Mo[3] - M * cx * cx;
    const float Cyy = Mo[4] - M * cy * cy;
    const float Czz = Mo[5] - M * cz * cz;
    const float Cxy = Mo[6] - M * cx * cy;
    const float Cxz = Mo[7] - M * cx * cz;
    const float Cyz = Mo[8] - M * cy * cz;

    const float Ixx = Cyy + Czz, Iyy = Cxx + Czz, Izz = Cxx + Cyy;
    const float Ixy = -Cxy, Ixz = -Cxz, Iyz = -Cyz;

    float* I = out_inertia + b * 9;
    I[0] = Ixx; I[1] = Ixy; I[2] = Ixz;
    I[3] = Ixy; I[4] = Iyy; I[5] = Iyz;
    I[6] = Ixz; I[7] = Iyz; I[8] = Izz;
}

extern "C" void kernel_launch(void* const* d_in, const int* in_sizes, int n_in,
                              void* d_out, int out_size, void* d_ws, size_t ws_size,
                              hipStream_t stream) {
    const float* angles  = (const float*)d_in[0];
    const float* points  = (const float*)d_in[1];
    const float* origins = (const float*)d_in[2];
    const float* joint_w = (const float*)d_in[3];
    const float* base_w  = (const float*)d_in[4];
    const float* trans   = (const float*)d_in[5];
    const float* rot     = (const float*)d_in[6];
    const float* mass    = (const float*)d_in[7];

    float* world   = (float*)d_out;                      // (B,N,3)
    float* com     = world + (size_t)B_DIM * N_DIM * 3;  // (B,3)
    float* inertia = com + B_DIM * 3;                    // (B,3,3)

    float* moments_partial = (float*)d_ws;               // B*64*10 floats = 40 KB

    dim3 grid(BLOCKS_PER_BATCH, B_DIM);
    skin_moments_kernel<<<grid, THREADS, 0, stream>>>(
        angles, points, origins, joint_w, base_w, trans, rot, mass,
        world, moments_partial);
    finalize_kernel<<<1, 32, 0, stream>>>(moments_partial, com, inertia);
}